// OrthogonalLayer1D_21723944583515
// MI455X (gfx1250) — compile-verified
//
#include <hip/hip_runtime.h>

typedef float v2f __attribute__((ext_vector_type(2)));
typedef float v8f __attribute__((ext_vector_type(8)));
typedef int   v4i __attribute__((ext_vector_type(4)));

#define NM   16          // models (vectors per batch)
#define DIMV 512         // vector dimension
#define NS   16384       // batches
#define LDA  (DIMV + 4)  // row pad: 516 floats -> 16B-aligned rows, 4-bank rotation

#define AS1 __attribute__((address_space(1)))
#define AS3 __attribute__((address_space(3)))

typedef AS1 v4i* gvec_t;   // global 128-bit pointer
typedef AS3 v4i* lvec_t;   // LDS    128-bit pointer

#if defined(__has_builtin)
#  if __has_builtin(__builtin_amdgcn_global_load_async_to_lds_b128)
#    define HAS_ASYNC_LD 1
#  else
#    define HAS_ASYNC_LD 0
#  endif
#  if __has_builtin(__builtin_amdgcn_global_store_async_from_lds_b128)
#    define HAS_ASYNC_ST 1
#  else
#    define HAS_ASYNC_ST 0
#  endif
#else
#  define HAS_ASYNC_LD 0
#  define HAS_ASYNC_ST 0
#endif

__device__ __forceinline__ void wait_async0() {
#if defined(__has_builtin) && __has_builtin(__builtin_amdgcn_s_wait_asynccnt)
    __builtin_amdgcn_s_wait_asynccnt(0);
#else
    asm volatile("s_wait_asynccnt 0" ::: "memory");
#endif
}

__launch_bounds__(32)
__global__ void cholqr_gs_kernel(const float* __restrict__ x, float* __restrict__ out)
{
    __shared__ float As[NM][LDA];   // A tile (and later Q tile), 33 KB
    __shared__ float Gm[NM][18];    // Gram -> Cholesky L (in place)
    __shared__ float Wm[NM][18];    // W = L^{-1}

    const int lane = threadIdx.x;                    // 0..31, one wave per block
    const long long b  = blockIdx.x;                 // batch index
    const long long MS = (long long)NS * DIMV;       // stride between models

    // ---------------- load A (16 x 512): async global->LDS if available -----
#if HAS_ASYNC_LD
    #pragma unroll 8
    for (int it = 0; it < (NM * DIMV / 4) / 32; ++it) {     // 64 issues x 512B
        int idx = it * 32 + lane;
        int row = idx >> 7;                                  // / 128
        int c4  = idx & 127;                                 // % 128
        const float* gp = x + (long long)row * MS + b * DIMV + (long long)c4 * 4;
        __builtin_amdgcn_global_load_async_to_lds_b128(
            (gvec_t)gp, (lvec_t)&As[row][c4 * 4], 0, 0);
    }
    wait_async0();
#else
    #pragma unroll 8
    for (int it = 0; it < (NM * DIMV / 4) / 32; ++it) {
        int idx = it * 32 + lane;
        int row = idx >> 7;
        int c4  = idx & 127;
        float4 v = *(const float4*)(x + (long long)row * MS + b * DIMV + (long long)c4 * 4);
        *(float4*)&As[row][c4 * 4] = v;
    }
#endif
    __syncthreads();

    const int mrow  = lane & 15;            // M (and N) index of this lane
    const int khalf = (lane >> 4) * 2;      // K sub-offset: 0 or 2

    // ---------------- Gram: G = A * A^T via V_WMMA_F32_16X16X4_F32 ----------
    // A-frag (16x4 f32): lane L holds A[L%16][k0 + 2*(L/16) + j], j = vgpr 0/1
    // B-frag (4x16 f32) of A^T has the identical per-lane values -> reuse regs.
    v8f g = {0.f, 0.f, 0.f, 0.f, 0.f, 0.f, 0.f, 0.f};
    #pragma unroll 4
    for (int k0 = 0; k0 < DIMV; k0 += 4) {
        v2f a = *(const v2f*)&As[mrow][k0 + khalf];
        g = __builtin_amdgcn_wmma_f32_16x16x4_f32(false, a, false, a,
                                                  (short)0, g, false, false);
    }
    // D layout: vgpr v, lane L -> G[v + 8*(L/16)][L%16]
    {
        int rb = (lane >> 4) * 8;
        #pragma unroll
        for (int v = 0; v < 8; ++v)
            Gm[rb + v][mrow] = g[v];
    }
    __syncthreads();

    // ---------------- Cholesky G = L L^T (in place, lower triangle) ---------
    for (int j = 0; j < NM; ++j) {
        if (lane == j) Gm[j][j] = sqrtf(Gm[j][j]);
        __syncthreads();
        if (lane < NM && lane > j) Gm[lane][j] /= Gm[j][j];
        __syncthreads();
        if (lane < NM && lane > j) {
            float lij = Gm[lane][j];
            for (int k = j + 1; k <= lane; ++k)
                Gm[lane][k] -= lij * Gm[k][j];
        }
        __syncthreads();
    }

    // ---------------- W = L^{-1} (lane c computes column c) ----------------
    if (lane < NM) {
        const int c = lane;
        float w[NM];
        #pragma unroll
        for (int i = 0; i < NM; ++i) {
            float s = (i == c) ? 1.0f : 0.0f;
            #pragma unroll
            for (int k = 0; k < NM; ++k)
                if (k < i) s -= Gm[i][k] * w[k];   // w[k]==0 for k<c
            w[i] = (i >= c) ? s / Gm[i][i] : 0.0f;
        }
        #pragma unroll
        for (int i = 0; i < NM; ++i) Wm[i][c] = w[i];
    }
    __syncthreads();

    // ---------------- Q = W * A, 16-wide column chunks, in-place into As ----
    const int rbase = (lane >> 4) * 8;
    for (int c0 = 0; c0 < DIMV; c0 += 16) {
        v8f q = {0.f, 0.f, 0.f, 0.f, 0.f, 0.f, 0.f, 0.f};
        #pragma unroll
        for (int kc = 0; kc < NM; kc += 4) {
            // A-matrix = W: lane L -> W[L%16][kc + 2*(L/16) + j]
            v2f wfrag = *(const v2f*)&Wm[mrow][kc + khalf];
            // B-matrix = A[kc:kc+4, c0:c0+16]: lane L, vgpr j ->
            //   As[kc + 2*(L/16) + j][c0 + L%16]
            v2f bfrag;
            bfrag.x = As[kc + khalf + 0][c0 + mrow];
            bfrag.y = As[kc + khalf + 1][c0 + mrow];
            q = __builtin_amdgcn_wmma_f32_16x16x4_f32(false, wfrag, false, bfrag,
                                                      (short)0, q, false, false);
        }
        // write Q chunk back over As columns c0..c0+15 (only read by this chunk)
        #pragma unroll
        for (int v = 0; v < 8; ++v)
            As[rbase + v][c0 + mrow] = q[v];
    }
    __syncthreads();   // all ds writes of Q complete (DScnt) before LDS->global

    // ---------------- store Q (16 x 512): async LDS->global if available ----
#if HAS_ASYNC_ST
    #pragma unroll 8
    for (int it = 0; it < (NM * DIMV / 4) / 32; ++it) {
        int idx = it * 32 + lane;
        int row = idx >> 7;
        int c4  = idx & 127;
        float* gp = out + (long long)row * MS + b * DIMV + (long long)c4 * 4;
        __builtin_amdgcn_global_store_async_from_lds_b128(
            (gvec_t)gp, (lvec_t)&As[row][c4 * 4], 0, 0);
    }
    wait_async0();
#else
    #pragma unroll 8
    for (int it = 0; it < (NM * DIMV / 4) / 32; ++it) {
        int idx = it * 32 + lane;
        int row = idx >> 7;
        int c4  = idx & 127;
        *(float4*)(out + (long long)row * MS + b * DIMV + (long long)c4 * 4) =
            *(const float4*)&As[row][c4 * 4];
    }
#endif
}

extern "C" void kernel_launch(void* const* d_in, const int* in_sizes, int n_in,
                              void* d_out, int out_size, void* d_ws, size_t ws_size,
                              hipStream_t stream) {
    const float* x = (const float*)d_in[0];
    float* out = (float*)d_out;
    (void)in_sizes; (void)n_in; (void)out_size; (void)d_ws; (void)ws_size;
    hipLaunchKernelGGL(cholqr_gs_kernel, dim3(NS), dim3(32), 0, stream, x, out);
}